// BoundaryPredictor2_56229711839294
// MI455X (gfx1250) — compile-verified
//
#include <hip/hip_runtime.h>

#define B_ 8
#define L_ 8192
#define D_ 256
#define TILE 128
#define EPS_ 1.1920929e-07f

typedef __attribute__((ext_vector_type(2))) float v2f;
typedef __attribute__((ext_vector_type(8))) float v8f;

// ---------------------------------------------------------------------------
// K1: M[d][e] = sum_c Wq[c][d] * Wk[c][e]   (M = Wq^T Wk, 256x256)
// Also zero the global reduction scalars (must be re-zeroed every call).
// ---------------------------------------------------------------------------
__global__ __launch_bounds__(256) void k_gram(const float* __restrict__ Wq,
                                              const float* __restrict__ Wk,
                                              float* __restrict__ M,
                                              float* __restrict__ scal) {
  const int d = blockIdx.x;
  const int e = threadIdx.x;
  if (blockIdx.x == 0 && e < 2) scal[e] = 0.0f;
  float acc = 0.0f;
  for (int c = 0; c < D_; ++c)
    acc = fmaf(Wq[c * D_ + d], Wk[c * D_ + e], acc);
  M[d * D_ + e] = acc;
}

// ---------------------------------------------------------------------------
// K2: fused normalize + Y = Hn[t0+1 .. t0+128] * M + rowwise dot with
//     Hn[t0 .. t0+127]  ->  cos_sim[b, t0 .. t0+127]
// 8 waves per block; wave w owns 16 output rows. f32 WMMA 16x16x4.
// Dynamic LDS: hn (129*256 f32) | mt (16*256 f32, col-tile of M transposed)
//              | cosl (128 f32)
// ---------------------------------------------------------------------------
__global__ __launch_bounds__(256) void k_cos(const float* __restrict__ hidden,
                                             const float* __restrict__ M,
                                             float* __restrict__ cosw) {
  extern __shared__ float lds[];
  float* hn   = lds;                     // (TILE+1) * D_
  float* mt   = hn + (TILE + 1) * D_;    // 16 * D_
  float* cosl = mt + 16 * D_;            // TILE

  const int tilesPerB = L_ / TILE;
  const int b    = blockIdx.x / tilesPerB;
  const int t0   = (blockIdx.x % tilesPerB) * TILE;
  const int tid  = threadIdx.x;
  const int lane = tid & 31;
  const int w    = tid >> 5;

  // --- normalize rows t0 .. t0+TILE into LDS (one wave per row) ---
  for (int r = w; r < TILE + 1; r += 8) {
    int tok = t0 + r;
    if (tok > L_ - 1) tok = L_ - 1;   // clamp; garbage row guarded at store
    const float4* src = (const float4*)(hidden + ((size_t)b * L_ + tok) * D_);
    float4 x0 = src[2 * lane + 0];
    float4 x1 = src[2 * lane + 1];
    float ss = x0.x * x0.x + x0.y * x0.y + x0.z * x0.z + x0.w * x0.w
             + x1.x * x1.x + x1.y * x1.y + x1.z * x1.z + x1.w * x1.w;
    for (int off = 16; off > 0; off >>= 1) ss += __shfl_xor(ss, off, 32);
    float sc = 1.0f / fmaxf(sqrtf(ss), 1e-12f);
    x0.x *= sc; x0.y *= sc; x0.z *= sc; x0.w *= sc;
    x1.x *= sc; x1.y *= sc; x1.z *= sc; x1.w *= sc;
    float4* dst = (float4*)(hn + r * D_);
    dst[2 * lane + 0] = x0;
    dst[2 * lane + 1] = x1;
  }
  if (tid < TILE) cosl[tid] = 0.0f;
  __syncthreads();

  const int half = lane >> 4;      // 0: lanes 0-15, 1: lanes 16-31
  const int l16  = lane & 15;
  const int rowA = 16 * w + l16;   // local output row this lane feeds (A matrix)
  const int koff = 2 * half;       // K offset per ISA A/B lane layout

  float acc[8];
#pragma unroll
  for (int r = 0; r < 8; ++r) acc[r] = 0.0f;

  for (int e0 = 0; e0 < D_; e0 += 16) {
    __syncthreads();  // previous mt readers done
    // stage mt[c][d] = M[d][e0+c]  (transposed so B operand is contiguous)
#pragma unroll 1
    for (int d0 = 0; d0 < D_; d0 += 16) {
      int d = d0 + (tid >> 4);
      int c = tid & 15;
      mt[c * D_ + d] = M[(size_t)d * D_ + e0 + c];
    }
    __syncthreads();

    v8f c8 = {0.f, 0.f, 0.f, 0.f, 0.f, 0.f, 0.f, 0.f};
#pragma unroll 8
    for (int k0 = 0; k0 < D_; k0 += 4) {
      // A (16x4, rows = hn[t0+1+rowA]): lanes<16 K={k0,k0+1}, lanes>=16 K={k0+2,k0+3}
      v2f a  = *(const v2f*)(hn + (rowA + 1) * D_ + k0 + koff);
      // B (4x16, cols of M tile): same K split, col = l16
      v2f bm = *(const v2f*)(mt + l16 * D_ + k0 + koff);
      c8 = __builtin_amdgcn_wmma_f32_16x16x4_f32(false, a, false, bm,
                                                 (short)0, c8, false, false);
    }
    // C layout: VGPR r, lanes 0-15 -> row r, lanes 16-31 -> row r+8; col = l16
#pragma unroll
    for (int r = 0; r < 8; ++r) {
      int rowc = 16 * w + r + 8 * half;
      acc[r] += c8[r] * hn[rowc * D_ + e0 + l16];
    }
  }
#pragma unroll
  for (int r = 0; r < 8; ++r)
    atomicAdd(&cosl[16 * w + r + 8 * half], acc[r]);   // ds_add_f32
  __syncthreads();

  if (tid < TILE) {
    int j = t0 + tid;
    if (j <= L_ - 2) cosw[(size_t)b * L_ + j] = cosl[tid];
  }
}

// ---------------------------------------------------------------------------
// K3: per batch row: boundary formula, forced last-real boundary, inclusive
// scan -> segment starts; writes shortened mask, n_seg, f; global scalars.
// ---------------------------------------------------------------------------
__global__ __launch_bounds__(256) void k_boundary(const float* __restrict__ cosw,
                                                  const float* __restrict__ noise,
                                                  const float* __restrict__ attn,
                                                  int* __restrict__ segw,
                                                  int* __restrict__ nsegw,
                                                  int* __restrict__ flenw,
                                                  float* __restrict__ scal,
                                                  float* __restrict__ outMask) {
  __shared__ float s[256];
  __shared__ int fSh;
  const int b = blockIdx.x;
  const int tid = threadIdx.x;

  // f = #real tokens (mask is contiguous ones then zeros)
  float cnt = 0.0f;
  for (int l = tid; l < L_; l += 256) cnt += attn[(size_t)b * L_ + l];
  s[tid] = cnt;
  __syncthreads();
  for (int off = 128; off > 0; off >>= 1) {
    if (tid < off) s[tid] += s[tid + off];
    __syncthreads();
  }
  if (tid == 0) fSh = (int)(s[0] + 0.5f);
  __syncthreads();
  const int f = fSh;
  __syncthreads();

  float carry = 0.0f;
  for (int c0 = 0; c0 < L_; c0 += 256) {
    int l = c0 + tid;
    float mk = attn[(size_t)b * L_ + l];
    float p;
    if (l == 0) {
      p = 1.0f;
    } else {
      float cs = cosw[(size_t)b * L_ + l - 1];
      p = (1.0f - cs) * 0.5f;
      p = fminf(fmaxf(p, 0.0f), 1.0f);
    }
    p = fminf(fmaxf(p, EPS_), 1.0f - EPS_);
    float logit = logf(p) - log1pf(-p);
    float u = fminf(fmaxf(noise[(size_t)b * L_ + l], EPS_), 1.0f - EPS_);
    float z = logit + logf(u) - log1pf(-u);          // TEMP = 1
    float soft = 1.0f / (1.0f + expf(-z));
    float hard = (soft > 0.5f) ? 1.0f : 0.0f;
    hard *= mk;
    if (f < L_ && l == f - 1) hard = 1.0f;           // forced last-real boundary

    // inclusive scan (Hillis-Steele)
    s[tid] = hard;
    __syncthreads();
    for (int off = 1; off < 256; off <<= 1) {
      float v = (tid >= off) ? s[tid - off] : 0.0f;
      __syncthreads();
      s[tid] += v;
      __syncthreads();
    }
    float cum = carry + s[tid];
    if (hard > 0.5f)
      segw[(size_t)b * L_ + ((int)(cum + 0.5f) - 1)] = l;
    float tot = s[255];
    __syncthreads();
    carry += tot;
  }

  int nb = (int)(carry + 0.5f);
  if (tid == 0) {
    nsegw[b] = nb;
    flenw[b] = f;
    atomicAdd(&scal[0], carry);
    atomicAdd(&scal[1], (float)f);
  }
  for (int l = tid; l < L_; l += 256)
    outMask[(size_t)b * L_ + l] = (l < nb) ? 1.0f : 0.0f;
}

// ---------------------------------------------------------------------------
// K4: segment-mean pooling; contiguous segment ranges; thread = channel.
// Writes zeros for rows >= n_seg (output must be fully rewritten each call).
// ---------------------------------------------------------------------------
__global__ __launch_bounds__(256) void k_pool(const float* __restrict__ hidden,
                                              const int* __restrict__ segw,
                                              const int* __restrict__ nsegw,
                                              const int* __restrict__ flenw,
                                              float* __restrict__ pooled) {
  const int rowsPer = 8;
  const int chunks = L_ / rowsPer;
  const int b  = blockIdx.x / chunks;
  const int s0 = (blockIdx.x % chunks) * rowsPer;
  const int d  = threadIdx.x;
  const int ns = nsegw[b];
  const int f  = flenw[b];
  for (int s = s0; s < s0 + rowsPer; ++s) {
    float val = 0.0f;
    if (s < ns) {
      int start = (s == 0) ? 0 : segw[(size_t)b * L_ + s];  // clip(seg,0,..)
      int end   = (s + 1 < ns) ? segw[(size_t)b * L_ + s + 1] : f;
      float sum = 0.0f;
      for (int l = start; l < end; ++l)
        sum += hidden[((size_t)b * L_ + l) * D_ + d];
      val = sum / ((float)(end - start) + 1e-9f);
    }
    pooled[((size_t)b * L_ + s) * D_ + d] = val;
  }
}

// ---------------------------------------------------------------------------
// K5: binomial NLL loss (Stirling lgamma, f64)
// ---------------------------------------------------------------------------
__device__ double lgamma_stirling(double x) {
  double acc = 0.0;
  while (x < 8.0) { acc -= log(x); x += 1.0; }
  double inv = 1.0 / x, inv2 = inv * inv;
  double ser = inv * (8.3333333333333333e-2
             - inv2 * (2.7777777777777778e-3
             - inv2 * 7.9365079365079365e-4));
  return (x - 0.5) * log(x) - x + 0.91893853320467274 + ser + acc;
}

__global__ void k_loss(const float* __restrict__ scal, float* __restrict__ out3) {
  double k = (double)scal[0];
  double n = (double)scal[1];
  const double p = 0.2;
  double lp = lgamma_stirling(n + 1.0) - lgamma_stirling(k + 1.0)
            - lgamma_stirling(n - k + 1.0)
            + k * log(p) + (n - k) * log1p(-p);
  out3[0] = (float)(-lp / n);
  out3[1] = (float)k;
  out3[2] = (float)n;
}

// ---------------------------------------------------------------------------
extern "C" void kernel_launch(void* const* d_in, const int* in_sizes, int n_in,
                              void* d_out, int out_size, void* d_ws, size_t ws_size,
                              hipStream_t stream) {
  (void)in_sizes; (void)n_in; (void)out_size; (void)ws_size;
  const float* hidden = (const float*)d_in[0];
  const float* Wq     = (const float*)d_in[1];
  const float* Wk     = (const float*)d_in[2];
  const float* noise  = (const float*)d_in[3];
  const float* attn   = (const float*)d_in[4];

  float* out     = (float*)d_out;
  float* pooled  = out;                              // B*L*D
  float* out3    = out + (size_t)B_ * L_ * D_;       // loss, num_boundaries, total_positions
  float* outMask = out3 + 3;                         // B*L

  float* wsf   = (float*)d_ws;
  float* Mw    = wsf;                                // 256*256
  float* cosw  = wsf + D_ * D_;                      // B*L
  int*   segw  = (int*)(cosw + (size_t)B_ * L_);     // B*L
  int*   nsegw = segw + (size_t)B_ * L_;             // B
  int*   flenw = nsegw + B_;                         // B
  float* scal  = (float*)(flenw + B_);               // 2

  k_gram<<<D_, D_, 0, stream>>>(Wq, Wk, Mw, scal);

  size_t ldsBytes = ((size_t)(TILE + 1) * D_ + 16 * D_ + TILE) * sizeof(float);
  k_cos<<<B_ * (L_ / TILE), 256, ldsBytes, stream>>>(hidden, Mw, cosw);

  k_boundary<<<B_, 256, 0, stream>>>(cosw, noise, attn, segw, nsegw, flenw,
                                     scal, outMask);
  k_pool<<<B_ * (L_ / 8), 256, 0, stream>>>(hidden, segw, nsegw, flenw, pooled);
  k_loss<<<1, 1, 0, stream>>>(scal, out3);
}